// MySoftArgmax2D_14181982011982
// MI455X (gfx1250) — compile-verified
//
#include <hip/hip_runtime.h>
#include <hip/hip_bf16.h>

#define NCH 11
#define HH 2048
#define WW 2048
#define BETA 100.0f

typedef float v2f __attribute__((ext_vector_type(2)));
typedef float v8f __attribute__((ext_vector_type(8)));

// Layout-robust wave32 sum-reduction using V_WMMA_F32_16X16X4_F32:
// A = 0.5 in all 16x4 slots, B = partial value replicated in both B VGPRs.
// D[m][n] = 0.5 * sum_k B[k][n]; summing D's 16 columns = 0.5 * (sum of all
// 64 physical B slots) = 0.5 * 2 * (wave sum) = wave sum, regardless of the
// exact K<->(vgpr,lane-half) striping.
__device__ __forceinline__ float wave_sum_wmma(float p) {
    v2f a; a.x = 0.5f; a.y = 0.5f;
    v2f b; b.x = p;    b.y = p;
    v8f c = {};
    v8f d = __builtin_amdgcn_wmma_f32_16x16x4_f32(false, a, false, b,
                                                  (short)0, c, false, false);
    float s = d[0];
    #pragma unroll
    for (int off = 8; off >= 1; off >>= 1)
        s += __shfl_xor(s, off, 32);
    return s;
}

// Pass 1: per-row stats (row max M of beta*h, S = sum exp(beta*h - M),
// SX = sum exp(beta*h - M) * x). One wave per row, 64 f32 per lane in regs.
__global__ __launch_bounds__(256) void rowstats_kernel(
    const float* __restrict__ hm, float4* __restrict__ stats) {
    const int lane = threadIdx.x & 31;
    const int wv   = threadIdx.x >> 5;
    const int row  = blockIdx.x * 8 + wv;            // 0 .. NCH*HH-1 exactly
    const float4* p = (const float4*)(hm + (size_t)row * WW);

    float4 v[16];
    #pragma unroll
    for (int j = 0; j < 16; ++j) {
        float4 t = p[j * 32 + lane];
        v[j] = make_float4(t.x * BETA, t.y * BETA, t.z * BETA, t.w * BETA);
    }

    float mloc = v[0].x;
    #pragma unroll
    for (int j = 0; j < 16; ++j)
        mloc = fmaxf(mloc, fmaxf(fmaxf(v[j].x, v[j].y), fmaxf(v[j].z, v[j].w)));
    #pragma unroll
    for (int off = 16; off >= 1; off >>= 1)
        mloc = fmaxf(mloc, __shfl_xor(mloc, off, 32));

    float s = 0.f, sx = 0.f;
    #pragma unroll
    for (int j = 0; j < 16; ++j) {
        const float c0 = (float)(j * 128 + lane * 4);
        float e0 = __expf(v[j].x - mloc);
        float e1 = __expf(v[j].y - mloc);
        float e2 = __expf(v[j].z - mloc);
        float e3 = __expf(v[j].w - mloc);
        s += (e0 + e1) + (e2 + e3);
        sx = fmaf(e0, c0,        sx);
        sx = fmaf(e1, c0 + 1.f,  sx);
        sx = fmaf(e2, c0 + 2.f,  sx);
        sx = fmaf(e3, c0 + 3.f,  sx);
    }

    float S  = wave_sum_wmma(s);
    float SX = wave_sum_wmma(sx);

    if (lane == 0)
        stats[row] = make_float4(mloc, S, SX, 0.0f);
}

// Combine row stats over band [r_lo, r_hi]; rows outside contribute the
// analytic all-zeros term (value 0 per element before softmax).
__device__ __forceinline__ void combine_band(const float4* st, int r_lo, int r_hi,
                                             int lane, float& ax, float& ay) {
    float mloc = -INFINITY;
    for (int r = r_lo + lane; r <= r_hi; r += 32)
        mloc = fmaxf(mloc, st[r].x);
    #pragma unroll
    for (int off = 16; off >= 1; off >>= 1)
        mloc = fmaxf(mloc, __shfl_xor(mloc, off, 32));

    const int n_in  = (r_hi >= r_lo) ? (r_hi - r_lo + 1) : 0;
    const int n_out = HH - n_in;
    float M = mloc;
    if (n_out > 0) M = fmaxf(M, 0.0f);   // also fixes n_in==0 (mloc=-inf)

    float S = 0.f, SX = 0.f, SY = 0.f;
    for (int r = r_lo + lane; r <= r_hi; r += 32) {
        float4 v = st[r];
        float w = __expf(v.x - M);
        float sw = v.y * w;
        S  += sw;
        SX += v.z * w;
        SY = fmaf(sw, (float)r, SY);
    }
    #pragma unroll
    for (int off = 16; off >= 1; off >>= 1) {
        S  += __shfl_xor(S,  off, 32);
        SX += __shfl_xor(SX, off, 32);
        SY += __shfl_xor(SY, off, 32);
    }

    const float e0        = __expf(0.0f - M);
    const float sum_all_r = 0.5f * (float)HH * (float)(HH - 1);
    const float sum_in_r  = (n_in > 0) ? 0.5f * (float)(r_lo + r_hi) * (float)n_in : 0.0f;
    const float sum_out_r = sum_all_r - sum_in_r;

    const float denom = S  + e0 * (float)WW * (float)n_out;
    const float sxo   = e0 * (0.5f * (float)WW * (float)(WW - 1)) * (float)n_out;
    const float syo   = e0 * (float)WW * sum_out_r;

    ax = (SX + sxo) / denom;
    ay = (SY + syo) / denom;
}

// Pass 2: one wave. Full-map soft-argmax for ch 9,10, then the 9-step
// sequential band recurrence, then write the 22 outputs.
__global__ __launch_bounds__(32) void finalize_kernel(
    const float4* __restrict__ stats, float* __restrict__ out) {
    __shared__ float4 st[HH];
    const int lane = threadIdx.x;

    float axv[NCH], ayv[NCH];

    #pragma unroll
    for (int ch = 9; ch <= 10; ++ch) {
        const float4* base = stats + ch * HH;
        for (int r = lane; r < HH; r += 32) st[r] = base[r];
        __syncthreads();
        combine_band(st, 0, HH - 1, lane, axv[ch], ayv[ch]);
        __syncthreads();
    }

    float dis = ayv[10] - ayv[9];
    float dis_sum = 0.f, dis_num = 0.f;

    #pragma unroll
    for (int i = 8; i >= 0; --i) {
        const float y1 = ayv[i + 1];
        const float y2 = ayv[i + 2];
        const float last_y  = floorf(y1);
        const float tmp_dis = ceilf(y2 - y1);
        if (fabsf(tmp_dis - dis) > 0.35f * dis) {
            dis_sum += tmp_dis;
            dis_num += 1.0f;
            dis = dis_sum / fmaxf(dis_num, 1.0f);
        }
        const float start_raw = last_y - 1.8f * dis;
        const float end_y   = rintf(start_raw + 0.9f * 2.0f * dis);  // RNE == jnp.round
        const float start_y = rintf(start_raw);

        const float rlof = fminf(fmaxf(ceilf(start_y), 0.0f), 2048.0f);
        const float rhif = fminf(fmaxf(floorf(end_y), -1.0f), 2047.0f);
        const int r_lo = (int)rlof;
        const int r_hi = (int)rhif;

        const float4* base = stats + i * HH;
        for (int r = lane; r < HH; r += 32) st[r] = base[r];
        __syncthreads();
        combine_band(st, r_lo, r_hi, lane, axv[i], ayv[i]);
        __syncthreads();
    }

    if (lane == 0) {
        #pragma unroll
        for (int c = 0; c < NCH; ++c) {
            out[2 * c + 0] = axv[c];
            out[2 * c + 1] = ayv[c];
        }
    }
}

extern "C" void kernel_launch(void* const* d_in, const int* in_sizes, int n_in,
                              void* d_out, int out_size, void* d_ws, size_t ws_size,
                              hipStream_t stream) {
    (void)in_sizes; (void)n_in; (void)out_size; (void)ws_size;
    const float* hm  = (const float*)d_in[0];
    float*       out = (float*)d_out;
    float4*      stats = (float4*)d_ws;   // NCH*HH float4 = 352 KB scratch

    const int rows = NCH * HH;            // 22528, one wave per row
    rowstats_kernel<<<rows / 8, 256, 0, stream>>>(hm, stats);
    finalize_kernel<<<1, 32, 0, stream>>>(stats, out);
}